// GCNNalpha_25091198943904
// MI455X (gfx1250) — compile-verified
//
#include <hip/hip_runtime.h>
#include <math.h>

// ---------------------------------------------------------------------------
// GCNN-alpha forward for MI455X (gfx1250, wave32, WMMA).
// One workgroup per graph (B=1024). 256 threads = 8 waves; wave w owns the
// node strip [16w, 16w+16). All 128x128x128 matmuls use
// v_wmma_f32_16x16x4_f32 (f32 A/B/C: exact precision match vs f32 ref).
//
// Storage is chosen so WMMA fragments are K-contiguous and loaded as single
// ds_load_b64 with 16-bit immediate offsets (no per-stream address regs):
//   - S  : row-major  (A operand of S@h)
//   - h  : TRANSPOSED only, HT[f][i]  (B operand of S@h; strip-A of h@W)
//   - W  : staged TRANSPOSED, WT[o][f] (B operand of h@W)
// Strip stores go out as ds_store_b128 (8 consecutive floats per lane/tile).
// ---------------------------------------------------------------------------

typedef float v2f __attribute__((ext_vector_type(2)));
typedef float v4f __attribute__((ext_vector_type(4)));
typedef float v8f __attribute__((ext_vector_type(8)));

#define NPG   128          // nodes per graph
#define FD    128          // feature dim
#define HOPS  3
#define OUTD  32
#define EH    1024         // directed edges per graph per half (NPG*DEG/2)
#define LD    132          // padded LDS row stride (floats): stride-4 banks
#define TSZ   (NPG * LD)   // floats per padded 128x128 matrix

__device__ __forceinline__ v8f wmma_f32(v2f a, v2f b, v8f c) {
  return __builtin_amdgcn_wmma_f32_16x16x4_f32(
      /*neg_a=*/false, a, /*neg_b=*/false, b,
      /*c_mod=*/(short)0, c, /*reuse_a=*/false, /*reuse_b=*/false);
}

// Generic 16x128 @ 128x128 strip-matmul, K = 128.
//   A element for lane: row (i0 + lane&15), k = kk + 2*(lane>>4) + {0,1};
//     aBase is lane-resolved; advances aStep floats per kk-step; pair offset
//     aPair floats (1 = contiguous b64, LD = strided pair).
//   B operand supplied TRANSPOSED (BT[n][k], row stride LD): per kk-step the
//     8 tj-streams are immediate offsets off one base -> ds_load_b64.
__device__ __forceinline__ void mm_strip(const float* aBase, int aStep, int aPair,
                                         const float* BT, v8f acc[8], int lane) {
  const int n  = lane & 15;
  const int kh = (lane >> 4) << 1;
  const float* bBase = BT + n * LD + kh;        // advances 4 floats per kk-step
  for (int kk = 0; kk < FD; kk += 4) {
    v2f a;
    if (aPair == 1) {
      a = *reinterpret_cast<const v2f*>(aBase + kk * aStep);
    } else {
      a.x = aBase[kk * aStep];
      a.y = aBase[kk * aStep + aPair];
    }
    const float* bk = bBase + kk;
#pragma unroll
    for (int tj = 0; tj < 8; ++tj) {
      v2f b = *reinterpret_cast<const v2f*>(bk + tj * (16 * LD));
      acc[tj] = wmma_f32(a, b, acc[tj]);
    }
  }
}

// C/D tile layout: lane L -> col n = L&15, rows M = v + 8*(L>>4).
// Per (tj, lane) the 8 result rows are CONTIGUOUS in HT -> two b128 stores.
__device__ __forceinline__ void store_strip_T(float* HT, const v8f t[8], int i0, int lane) {
  const int half = lane >> 4, n = lane & 15;
#pragma unroll
  for (int tj = 0; tj < 8; ++tj) {
    float* p = HT + (tj * 16 + n) * LD + i0 + 8 * half;
    v4f lo, hi;
    lo.x = t[tj][0]; lo.y = t[tj][1]; lo.z = t[tj][2]; lo.w = t[tj][3];
    hi.x = t[tj][4]; hi.y = t[tj][5]; hi.z = t[tj][6]; hi.w = t[tj][7];
    *reinterpret_cast<v4f*>(p)     = lo;
    *reinterpret_cast<v4f*>(p + 4) = hi;
  }
}

__global__ __launch_bounds__(256) void gcnn_alpha_kernel(
    const float* __restrict__ X,  const float* __restrict__ W1,
    const float* __restrict__ b1, const float* __restrict__ W2,
    const float* __restrict__ b2, const float* __restrict__ Wout,
    const float* __restrict__ bout, const float* __restrict__ alphap,
    const int* __restrict__ edge, float* __restrict__ out, int Ehalf) {
  extern __shared__ float smem[];
  float* S      = smem;                 // GSO, row-major          128x132
  float* HTa    = smem + 1 * TSZ;       // h^T ping                128x132
  float* HTb    = smem + 2 * TSZ;       // h^T pong                128x132
  float* WT     = smem + 3 * TSZ;       // W[k]^T staged           128x132
  float* pooled = smem + 4 * TSZ;       // 128
  float* r1     = pooled + NPG;         // 128
  float* r2     = r1 + NPG;             // 128

  const int g    = blockIdx.x;
  const int tid  = threadIdx.x;
  const int lane = tid & 31;
  const int i0   = (tid >> 5) * 16;     // wave's node-strip base
  const int m    = lane & 15;
  const int kh   = (lane >> 4) << 1;

  // ---- stage X[g]^T into HTa; zero adjacency ----
  const float* Xg = X + (size_t)g * NPG * FD;
  for (int idx = tid; idx < NPG * FD; idx += 256) {
    int i = idx >> 7, f = idx & 127;
    HTa[f * LD + i] = Xg[idx];
  }
  for (int idx = tid; idx < TSZ; idx += 256) S[idx] = 0.0f;
  __syncthreads();

  // ---- adjacency: graph g edges at [g*EH,(g+1)*EH) and [Ehalf+g*EH, ...) ----
  const int* row0 = edge;
  const int* row1 = edge + 2 * Ehalf;
  for (int e = tid; e < 2 * EH; e += 256) {
    int p = (e < EH) ? (g * EH + e) : (Ehalf + g * EH + (e - EH));
    int u = row0[p] & (NPG - 1);
    int v = row1[p] & (NPG - 1);
    S[u * LD + v] = 1.0f;               // .set semantics: duplicate-safe
  }
  __syncthreads();

  // ---- hub-normalized GSO: S = ds^(-a) A ds^(a-1), ds = max(deg,1) ----
  const float alpha = alphap[0];
  if (tid < NPG) {
    float d = 0.f;
    for (int j = 0; j < NPG; ++j) d += S[tid * LD + j];
    float ds = d > 0.f ? d : 1.f;
    r1[tid] = powf(ds, -alpha);
    r2[tid] = powf(ds, alpha - 1.0f);
  }
  __syncthreads();
  for (int idx = tid; idx < NPG * NPG; idx += 256) {
    int i = idx >> 7, j = idx & 127;
    S[i * LD + j] *= r1[i] * r2[j];
  }

  // ---- two layers: h = relu( sum_k (S^k h) W[k] + b ) ----
  const float* Wl[2] = {W1, W2};
  const float* bl[2] = {b1, b2};
  for (int layer = 0; layer < 2; ++layer) {
    v8f acc[8];
#pragma unroll
    for (int t = 0; t < 8; ++t) acc[t] = (v8f)0.0f;
    float* curT = HTa;
    float* nxtT = HTb;
    for (int k = 0; k <= HOPS; ++k) {
      __syncthreads();                  // prev-iter consumers of WT/nxtT done
      const float* Wk = Wl[layer] + (size_t)k * FD * FD;
      for (int idx = tid; idx < FD * FD; idx += 256) {
        int f = idx >> 7, o = idx & 127;
        WT[o * LD + f] = Wk[idx];       // stage W[k] transposed
      }
      __syncthreads();
      // acc += h_k @ W[k]: A-frag = curT[kk+kh][i0+m] (strided pair, 1 per 8 wmma)
      mm_strip(curT + kh * LD + i0 + m, /*aStep=*/LD, /*aPair=*/LD,
               WT, acc, lane);
      if (k < HOPS) {
        v8f prop[8];
#pragma unroll
        for (int t = 0; t < 8; ++t) prop[t] = (v8f)0.0f;
        // h_{k+1} = S @ h_k: A-frag = S[i0+m][kk+kh] contiguous b64
        mm_strip(S + (i0 + m) * LD + kh, /*aStep=*/1, /*aPair=*/1,
                 curT, prop, lane);
        store_strip_T(nxtT, prop, i0, lane);
        float* tmp = curT; curT = nxtT; nxtT = tmp;
      }
    }
    // bias + relu -> HTa (holds dead h2; k=3 top barrier covered its readers)
    {
      const float* bb = bl[layer];
      v8f r[8];
      const int n = lane & 15;
#pragma unroll
      for (int tj = 0; tj < 8; ++tj) {
        float bv = bb[tj * 16 + n];
#pragma unroll
        for (int v = 0; v < 8; ++v) {
          float x = acc[tj][v] + bv;
          r[tj][v] = x > 0.f ? x : 0.f;
        }
      }
      store_strip_T(HTa, r, i0, lane);
    }
    __syncthreads();
  }

  // ---- global max pool over nodes (contiguous HT rows), then W_out ----
  if (tid < NPG) {
    float mmax = -3.402823466e38f;
    for (int i = 0; i < NPG; ++i) {
      float v = HTa[tid * LD + i];
      mmax = v > mmax ? v : mmax;
    }
    pooled[tid] = mmax;
  }
  __syncthreads();
  if (tid < OUTD) {
    float s = bout[tid];
    for (int f = 0; f < FD; ++f) s += pooled[f] * Wout[f * OUTD + tid];
    out[(size_t)g * OUTD + tid] = s;
  }
}

extern "C" void kernel_launch(void* const* d_in, const int* in_sizes, int n_in,
                              void* d_out, int out_size, void* d_ws, size_t ws_size,
                              hipStream_t stream) {
  const float* X     = (const float*)d_in[0];
  const float* W1    = (const float*)d_in[1];
  const float* b1    = (const float*)d_in[2];
  const float* W2    = (const float*)d_in[3];
  const float* b2    = (const float*)d_in[4];
  const float* Wout  = (const float*)d_in[5];
  const float* bout  = (const float*)d_in[6];
  const float* alpha = (const float*)d_in[7];
  // d_in[8] = batch (unused: equal-sized graphs)
  const int*   edge  = (const int*)d_in[9];

  const int nGraphs = in_sizes[0] / (NPG * FD);   // 1024
  const int Ehalf   = in_sizes[9] / 4;            // B * eh = 1,048,576

  const size_t smemBytes = (size_t)(4 * TSZ + 3 * NPG) * sizeof(float); // ~266 KB

  gcnn_alpha_kernel<<<nGraphs, 256, smemBytes, stream>>>(
      X, W1, b1, W2, b2, Wout, bout, alpha, edge, (float*)d_out, Ehalf);
}